// Mask_15015205666961
// MI455X (gfx1250) — compile-verified
//
#include <hip/hip_runtime.h>
#include <hip/hip_bf16.h>

// CA raster-scan mask propagation, 8 images of 224x224, 9 passes.
// One wave32 per image. State kept as bitmasks in LDS (7 bits/lane * 32 lanes
// = 224 columns). Row recurrence solved as a 3-state FSM with log-depth
// cross-lane function-composition scan (ds_bpermute). Input streamed
// HBM->LDS with CDNA5 async-to-LDS copies double-buffered on ASYNCcnt.

#define IMG_H 224
#define IMG_W 224
#define NPASS 9
#define LPW 7  // columns per lane (32*7 = 224)

__device__ __forceinline__ unsigned shfl32(unsigned v, int srcLane) {
  return (unsigned)__builtin_amdgcn_ds_bpermute(srcLane << 2, (int)v);
}
__device__ __forceinline__ int shfl32i(int v, int srcLane) {
  return __builtin_amdgcn_ds_bpermute(srcLane << 2, v);
}

// FSM over states {0,1,2} (trailing-ones run length, capped at 2).
// Function encoded in 6 bits: out(s) at bits [2s+1:2s].
__device__ __forceinline__ unsigned fsm_apply(unsigned f, unsigned s) {
  return (f >> (2u * s)) & 3u;
}
// returns g∘f (apply f first, then g)
__device__ __forceinline__ unsigned fsm_compose(unsigned g, unsigned f) {
  unsigned r0 = fsm_apply(g, f & 3u);
  unsigned r1 = fsm_apply(g, (f >> 2) & 3u);
  unsigned r2 = fsm_apply(g, (f >> 4) & 3u);
  return r0 | (r1 << 2) | (r2 << 4);
}

__global__ __launch_bounds__(32) void ca_mask_kernel(const float* __restrict__ x,
                                                     float* __restrict__ out) {
  __shared__ float stage[2][IMG_W];            // async staging (double buffer)
  __shared__ unsigned char cur[IMG_H * 32];    // pre-pass mask bits (7/lane)
  __shared__ unsigned char nxt[IMG_H * 32];    // updated (new) bits

  const int lane = threadIdx.x;                // 0..31
  const int img  = blockIdx.x;
  const float* src = x + (size_t)img * (IMG_H * IMG_W);
  float*       dst = out + (size_t)img * (IMG_H * IMG_W);

  // Flat address of LDS object: low 32 bits == LDS byte offset (shared aperture).
  const unsigned stage_off = (unsigned)(size_t)(&stage[0][0]);

  // ---------------- pass-0 threshold: stream rows via async-to-LDS ----------
  auto issue_row = [&](int h) {
    const float* rp = src + h * IMG_W;
    unsigned lb = stage_off + (unsigned)((h & 1) * IMG_W * 4);
#pragma unroll
    for (int j = 0; j < LPW; ++j) {
      unsigned laddr = lb + (unsigned)((j * 32 + lane) * 4);
      unsigned long long ga = (unsigned long long)(rp + j * 32 + lane);
      asm volatile("global_load_async_to_lds_b32 %0, %1, off"
                   :: "v"(laddr), "v"(ga) : "memory");
    }
  };

  issue_row(0);
  for (int h = 0; h < IMG_H; ++h) {
    if (h + 1 < IMG_H) {
      issue_row(h + 1);                              // overlap next row
      asm volatile("s_wait_asynccnt 7" ::: "memory"); // row h's 7 copies done
    } else {
      asm volatile("s_wait_asynccnt 0" ::: "memory");
    }
    unsigned bits = 0;
    const float* sp = &stage[h & 1][lane * LPW];
#pragma unroll
    for (int i = 0; i < LPW; ++i)
      bits |= (sp[i] > 0.8f ? 1u : 0u) << i;
    cur[h * 32 + lane] = (unsigned char)bits;
  }

  // ---------------- 9 CA passes on bitmasks ---------------------------------
  for (int pass = 0; pass < NPASS; ++pass) {
    int hmin = IMG_H, hmax = -1;
    unsigned colAny = 0;

    for (int h = 0; h < IMG_H; ++h) {
      unsigned c  = cur[h * 32 + lane];
      unsigned cn = (h + 1 < IMG_H) ? (unsigned)cur[(h + 1) * 32 + lane] : 0u;
      // below: h<222 -> c & c(h+1); h==222 -> c; h==223 -> 0
      unsigned below = (h < IMG_H - 2) ? (c & cn) : ((h == IMG_H - 2) ? c : 0u);
      // right: c & shift-left-neighborhood (col w+1); col222 -> c; col223 -> 0
      unsigned nb0 = shfl32(c, lane + 1) & 1u;       // lane L+1 bit0 (col 7L+7)
      unsigned right = c & ((c >> 1) | (nb0 << 6));
      if (lane == 31) right = (right & 0x1Fu) | (c & 0x20u); // patch cols 222/223
      // above from updated rows (nxt): h==0 -> 0; h==1 -> nxt[0]; else AND
      unsigned above;
      if (h == 0)      above = 0u;
      else if (h == 1) above = (unsigned)nxt[0 * 32 + lane];
      else             above = (unsigned)(nxt[(h - 1) * 32 + lane] &
                                          nxt[(h - 2) * 32 + lane]);
      unsigned base2 = below & right;
      unsigned base1 = (below ^ right) & 0x7Fu;
      unsigned base0 = (~(below | right)) & 0x7Fu;
      unsigned A = (base2 | (base1 & above)) & 0x7Fu;       // b >= 2
      unsigned B = ((base1 & ~above) | (base0 & above)) & 0x7Fu; // b == 1

      // Build per-lane transition function over its 7 columns.
      // id=0x24; fA(s)=min(s+1,2)=0x29; fB(s)=(s==2?2:0)=0x20; fZ=0x00
      unsigned F = 0x24u;
#pragma unroll
      for (int i = 0; i < LPW; ++i) {
        unsigned st = ((A >> i) & 1u) ? 0x29u : (((B >> i) & 1u) ? 0x20u : 0x00u);
        F = fsm_compose(st, F);
      }
      // Inclusive function-composition scan across lanes (Hillis-Steele).
      unsigned S = F;
#pragma unroll
      for (int d = 1; d < 32; d <<= 1) {
        unsigned g = shfl32(S, lane - d);
        if (lane >= d) S = fsm_compose(S, g);
      }
      // Entering state: s0=1 (fictitious val[-1]=1, val[-2]=0).
      unsigned E = shfl32(S, lane - 1);
      unsigned s = (lane == 0) ? 1u : fsm_apply(E, 1u);
      // Replay 7 columns emitting val bits.
      unsigned o = 0;
#pragma unroll
      for (int i = 0; i < LPW; ++i) {
        unsigned a = (A >> i) & 1u, b = (B >> i) & 1u;
        unsigned val = a | (b & (unsigned)(s == 2u));
        s = a ? ((s >= 1u) ? 2u : 1u) : (val ? 2u : 0u);
        o |= val << i;
      }
      nxt[h * 32 + lane] = (unsigned char)o;
      colAny |= o;
      unsigned anyb = __builtin_amdgcn_ballot_w32(o != 0u); // uniform
      if (anyb) { if (h < hmin) hmin = h; hmax = h; }
    }

    // Column bbox from new bits: wmin..wmax (wmax fill is EXCLUSIVE).
    int lmin = colAny ? (lane * LPW + __builtin_ctz(colAny)) : 0x7FFFFFFF;
    int lmax = colAny ? (lane * LPW + (31 - __builtin_clz(colAny))) : -1;
#pragma unroll
    for (int d = 16; d >= 1; d >>= 1) {
      int om = shfl32i(lmin, lane ^ d);
      int ox = shfl32i(lmax, lane ^ d);
      lmin = om < lmin ? om : lmin;
      lmax = ox > lmax ? ox : lmax;
    }
    int wmin = lmin, wmax = lmax;
    if (hmax < 0) { hmin = 0; hmax = IMG_H - 1; wmin = 0; wmax = IMG_W - 1; }

    // Per-lane rect mask for columns [wmin, wmax)  (wmax exclusive!)
    int lo = wmin - lane * LPW; if (lo < 0) lo = 0;
    int hi = wmax - lane * LPW; if (hi > LPW) hi = LPW;
    unsigned rmask = (hi > lo) ? ((((1u << (hi - lo)) - 1u) << lo) & 0x7Fu) : 0u;

    for (int h = 0; h < IMG_H; ++h) {
      unsigned v = nxt[h * 32 + lane];
      if (h >= hmin && h <= hmax) v |= rmask;
      cur[h * 32 + lane] = (unsigned char)v;
    }
  }

  // ---------------- write float output --------------------------------------
  for (int h = 0; h < IMG_H; ++h) {
    unsigned bits = cur[h * 32 + lane];
    float* op = dst + h * IMG_W + lane * LPW;
#pragma unroll
    for (int i = 0; i < LPW; ++i)
      op[i] = ((bits >> i) & 1u) ? 1.0f : 0.0f;
  }
}

extern "C" void kernel_launch(void* const* d_in, const int* in_sizes, int n_in,
                              void* d_out, int out_size, void* d_ws, size_t ws_size,
                              hipStream_t stream) {
  (void)n_in; (void)out_size; (void)d_ws; (void)ws_size;
  const float* x = (const float*)d_in[0];
  float* out = (float*)d_out;
  int imgs = in_sizes[0] / (IMG_H * IMG_W);   // B*C = 8
  ca_mask_kernel<<<imgs, 32, 0, stream>>>(x, out);
}